// AttentionBlock_60644938219597
// MI455X (gfx1250) — compile-verified
//
#include <hip/hip_runtime.h>
#include <math.h>

typedef __attribute__((ext_vector_type(2))) float v2f;
typedef __attribute__((ext_vector_type(4))) float f4;
typedef __attribute__((ext_vector_type(8))) float v8f;
typedef __attribute__((ext_vector_type(4))) int v4i;

constexpr int BB = 4;
constexpr int LL = 2048;
constexpr int DD = 1024;

#if defined(__gfx1250__) && __has_builtin(__builtin_amdgcn_global_load_async_to_lds_b128) && \
    __has_builtin(__builtin_amdgcn_global_load_async_to_lds_b32)
#define USE_ASYNC 1
#else
#define USE_ASYNC 0
#endif

#define AS1 __attribute__((address_space(1)))
#define AS3 __attribute__((address_space(3)))

__device__ __forceinline__ void async_wait_all() {
#if USE_ASYNC
#if __has_builtin(__builtin_amdgcn_s_wait_asynccnt)
  __builtin_amdgcn_s_wait_asynccnt(0);
#else
  asm volatile("s_wait_asynccnt 0" ::: "memory");
#endif
#endif
}

__device__ __forceinline__ void copy16(const float* g, float* l) {
#if USE_ASYNC
  __builtin_amdgcn_global_load_async_to_lds_b128((AS1 v4i*)(void*)g, (AS3 v4i*)(void*)l, 0, 0);
#else
  *(f4*)l = *(const f4*)g;
#endif
}

__device__ __forceinline__ void copy4(const float* g, float* l) {
#if USE_ASYNC
  __builtin_amdgcn_global_load_async_to_lds_b32((AS1 int*)(void*)g, (AS3 int*)(void*)l, 0, 0);
#else
  *l = *g;
#endif
}

// ---------------- fp32 WMMA GEMM: C[M,N] = A[M,K] * B[K,N] (+bias) ----------
// Tile: 128(M) x 64(N) per 256-thread block (8 wave32), TK = 32.
// Double-buffered LDS, async global->LDS staging (prefetch distance 1).
// BN_MAJOR=true : B source is stored [N][K] -> straight copy
// BN_MAJOR=false: B source is stored [K][N] -> transpose while staging (B32 async)
#define TM 128
#define TN 64
#define TK 32
#define ASTR 36
#define BSTR 36

template <bool BN_MAJOR, bool BIAS>
__global__ __launch_bounds__(256) void gemm_wmma_f32(
    const float* __restrict__ A, int lda, long long sA,
    const float* __restrict__ Bm, int ldb, long long sB,
    const float* __restrict__ bias,
    float* __restrict__ C, int ldc, long long sC, int Kdim) {
  __shared__ __align__(16) float Al[2][TM * ASTR];
  __shared__ __align__(16) float Bl[2][TN * BSTR];

  const int tid  = threadIdx.x;
  const int lane = tid & 31;
  const int wave = tid >> 5;          // 0..7 -> M sub-tile
  const int l16  = lane & 15;
  const int hsel = lane >> 4;         // 0/1 -> K half within fragment
  const int ntile = blockIdx.x * TN;
  const int mtile = blockIdx.y * TM;

  A  += (long long)blockIdx.z * sA + (long long)mtile * lda;
  Bm += (long long)blockIdx.z * sB;
  C  += (long long)blockIdx.z * sC;

  // issue the global->LDS copies for one K-chunk into buffer `buf`
  auto issue = [&](int buf, int k0) {
    // A tile: TM x TK, contiguous float4 copies
#pragma unroll
    for (int p = 0; p < 4; ++p) {
      int idx = tid + p * 256;                 // 1024 float4s
      int r = idx >> 3, c4 = (idx & 7) << 2;
      copy16(A + (long long)r * lda + k0 + c4, &Al[buf][r * ASTR + c4]);
    }
    if (BN_MAJOR) {
#pragma unroll
      for (int p = 0; p < 2; ++p) {
        int idx = tid + p * 256;               // 512 float4s
        int r = idx >> 3, c4 = (idx & 7) << 2;
        copy16(Bm + (long long)(ntile + r) * ldb + k0 + c4, &Bl[buf][r * BSTR + c4]);
      }
    } else {
#pragma unroll
      for (int p = 0; p < 8; ++p) {
        int idx = tid + p * 256;               // 2048 scalars, coalesced in n
        int kk = idx >> 6, n = idx & 63;
        copy4(Bm + (long long)(k0 + kk) * ldb + ntile + n, &Bl[buf][n * BSTR + kk]);
      }
    }
  };

  v8f acc0 = {}, acc1 = {}, acc2 = {}, acc3 = {};

  const int nchunks = Kdim / TK;
  issue(0, 0);
  for (int t = 0; t < nchunks; ++t) {
    const int cur = t & 1;
    async_wait_all();          // my async copies into buf[cur] are done
    __syncthreads();           // B1: buf[cur] visible to all; prev reads of buf[cur^1] done
    if (t + 1 < nchunks) issue(cur ^ 1, (t + 1) * TK);

    const float* ap = &Al[cur][(wave * 16 + l16) * ASTR + hsel * 2];
    const float* bp = &Bl[cur][l16 * BSTR + hsel * 2];
#pragma unroll
    for (int kk = 0; kk < TK; kk += 4) {
      v2f a  = *(const v2f*)(ap + kk);
      v2f b0 = *(const v2f*)(bp + kk + 0 * 16 * BSTR);
      v2f b1 = *(const v2f*)(bp + kk + 1 * 16 * BSTR);
      v2f b2 = *(const v2f*)(bp + kk + 2 * 16 * BSTR);
      v2f b3 = *(const v2f*)(bp + kk + 3 * 16 * BSTR);
      acc0 = __builtin_amdgcn_wmma_f32_16x16x4_f32(false, a, false, b0, (short)0, acc0, false, false);
      acc1 = __builtin_amdgcn_wmma_f32_16x16x4_f32(false, a, false, b1, (short)0, acc1, false, false);
      acc2 = __builtin_amdgcn_wmma_f32_16x16x4_f32(false, a, false, b2, (short)0, acc2, false, false);
      acc3 = __builtin_amdgcn_wmma_f32_16x16x4_f32(false, a, false, b3, (short)0, acc3, false, false);
    }
    __syncthreads();           // B2: all reads of buf[cur] done before it is refilled
  }

  // ---- writeback: VGPR v holds M = v + 8*hsel, N = l16 (16x16 f32 C layout)
  v8f accs[4] = {acc0, acc1, acc2, acc3};
#pragma unroll
  for (int nb = 0; nb < 4; ++nb) {
    int col = ntile + nb * 16 + l16;
    float bvv = BIAS ? bias[col] : 0.0f;
#pragma unroll
    for (int v = 0; v < 8; ++v) {
      int row = mtile + wave * 16 + v + hsel * 8;
      C[(long long)row * ldc + col] = accs[nb][v] + bvv;
    }
  }
}

// ---------------- softmax over axis=1 (query axis): per (b, j) column --------
__global__ __launch_bounds__(256) void colreduce_kernel(
    const float* __restrict__ raw, const int* __restrict__ ts,
    float* __restrict__ mcol, float* __restrict__ sinv) {
  int j = blockIdx.x * 256 + threadIdx.x;
  int b = blockIdx.y;
  const float* col = raw + (long long)b * LL * LL + j;
  const int* tsb = ts + b * LL;
  int tj = tsb[j];
  float m = -INFINITY, s = 0.0f;
  for (int i = 0; i < LL; ++i) {
    if (tsb[i] >= tj) {  // unmasked: ts[i] < ts[j] is masked out
      float l = col[(long long)i * LL] * 0.03125f;
      float mn = fmaxf(m, l);
      s = s * expf(m - mn) + expf(l - mn);
      m = mn;
    }
  }
  mcol[b * LL + j] = m;
  sinv[b * LL + j] = 1.0f / s;
}

__global__ __launch_bounds__(256) void rel_kernel(
    const float* __restrict__ raw, const int* __restrict__ ts,
    const float* __restrict__ mcol, const float* __restrict__ sinv,
    float* __restrict__ rel) {
  long long t = (long long)blockIdx.x * 256 + threadIdx.x;  // one per 4 j's
  int jq = (int)(t & (LL / 4 - 1));
  long long bi = t >> 9;               // b*L + i   (L/4 = 512)
  int b = (int)(bi >> 11);             // L = 2048
  int i = (int)(bi & (LL - 1));
  int j0 = jq << 2;
  int ti = ts[b * LL + i];
  long long base = bi * LL + j0;
  f4 r = *(const f4*)(raw + base);
  f4 o;
#pragma unroll
  for (int u = 0; u < 4; ++u) {
    int j = j0 + u;
    int tj = ts[b * LL + j];
    float val = 0.0f;
    if (ti >= tj)
      val = expf(r[u] * 0.03125f - mcol[b * LL + j]) * sinv[b * LL + j];
    o[u] = val;
  }
  *(f4*)(rel + base) = o;
}

__global__ __launch_bounds__(256) void copyx_kernel(const float* __restrict__ x,
                                                    float* __restrict__ out) {
  long long t = (long long)blockIdx.x * 256 + threadIdx.x;  // one float4
  int dq = (int)(t & 255);           // 1024/4 = 256 float4s per row
  long long bi = t >> 8;             // b*L + i
  *(f4*)(out + bi * 2048 + (dq << 2)) = *(const f4*)(x + bi * 1024 + (dq << 2));
}

// ---------------------------------------------------------------------------
extern "C" void kernel_launch(void* const* d_in, const int* in_sizes, int n_in,
                              void* d_out, int out_size, void* d_ws, size_t ws_size,
                              hipStream_t stream) {
  (void)in_sizes; (void)n_in; (void)out_size; (void)ws_size;
  const float* x  = (const float*)d_in[0];
  const int*   ts = (const int*)d_in[1];
  const float* Wq = (const float*)d_in[2];
  const float* bq = (const float*)d_in[3];
  const float* Wk = (const float*)d_in[4];
  const float* bk = (const float*)d_in[5];
  const float* Wv = (const float*)d_in[6];
  const float* bv = (const float*)d_in[7];

  float* out = (float*)d_out;                         // [B,L,2048]
  float* rel = out + (long long)BB * LL * 2048;       // [B,L,L]
  float* raw = rel + (long long)BB * LL * LL;         // [B,L,L]

  float* q  = (float*)d_ws;                          // [B*L, 1024]
  float* k  = q + (long long)BB * LL * DD;
  float* v  = k + (long long)BB * LL * DD;
  float* mb = v + (long long)BB * LL * DD;           // [B*L]
  float* sb = mb + BB * LL;                          // [B*L]

  dim3 blk(256);
  const long long sQKV = (long long)LL * DD;   // per-batch q/k/v stride
  const long long sRAW = (long long)LL * LL;
  const long long sOUT = (long long)LL * 2048;

  // 1-3) q/k/v = x @ W + b   (M=8192, N=1024, K=1024; W is [K][N] -> transpose stage)
  dim3 g1(DD / TN, (BB * LL) / TM, 1);
  gemm_wmma_f32<false, true><<<g1, blk, 0, stream>>>(x, DD, 0LL, Wq, DD, 0LL, bq, q, DD, 0LL, DD);
  gemm_wmma_f32<false, true><<<g1, blk, 0, stream>>>(x, DD, 0LL, Wk, DD, 0LL, bk, k, DD, 0LL, DD);
  gemm_wmma_f32<false, true><<<g1, blk, 0, stream>>>(x, DD, 0LL, Wv, DD, 0LL, bv, v, DD, 0LL, DD);

  // 4) raw = q @ k^T per batch (k rows are already N-major over j)
  dim3 g2(LL / TN, LL / TM, BB);
  gemm_wmma_f32<true, false><<<g2, blk, 0, stream>>>(q, DD, sQKV, k, DD, sQKV, nullptr,
                                                     raw, LL, sRAW, DD);

  // 5) per-(b,j) column max / inverse-sum over query axis with mask
  colreduce_kernel<<<dim3(LL / 256, BB), blk, 0, stream>>>(raw, ts, mb, sb);

  // 6) rel = masked softmax values
  rel_kernel<<<(unsigned)((long long)BB * LL * LL / 4 / 256), blk, 0, stream>>>(raw, ts, mb, sb, rel);

  // 7) attn = rel @ v per batch -> out[:, :, 1024:2048]
  dim3 g3(DD / TN, LL / TM, BB);
  gemm_wmma_f32<false, false><<<g3, blk, 0, stream>>>(rel, LL, sRAW, v, DD, sQKV, nullptr,
                                                      out + 1024, 2048, sOUT, LL);

  // 8) out[:, :, 0:1024] = x
  copyx_kernel<<<(unsigned)((long long)BB * LL * DD / 4 / 256), blk, 0, stream>>>(x, out);
}